// yolo_v3_77189152243795
// MI455X (gfx1250) — compile-verified
//
#include <hip/hip_runtime.h>
#include <cstdint>

// ---------------- problem constants (from reference) ----------------
#define A_NUM   3
#define D_NUM   85                  // 5 + 80 classes
#define C_NUM   (A_NUM * D_NUM)     // 255 channels
#define S_DIM   64
#define WH      (S_DIM * S_DIM)     // 4096
#define STRIDE_F 8.0f               // 512 / 64
#define TILE    64                  // spatial positions per workgroup
#define LDS_PITCH (TILE + 1)        // 65 floats: matches TDM pad (1 DW per 64 DW)
#define OUT_TILE (TILE * C_NUM)     // 16320 contiguous output floats per block

typedef uint32_t u32;
typedef __attribute__((ext_vector_type(4))) u32 u32x4;
typedef __attribute__((ext_vector_type(4))) int i32x4;
typedef __attribute__((ext_vector_type(8))) int i32x8;

__device__ __forceinline__ float sigm(float v) {
    // v_rcp_f32 (1 + e^-v)
    return __builtin_amdgcn_rcpf(1.0f + __expf(-v));
}

__global__ __launch_bounds__(256)
void yolo_v3_head_kernel(const float* __restrict__ x,
                         const float* __restrict__ anchors,
                         float* __restrict__ out) {
    // 255 channel rows, pitch 65 floats -> 66,300 B (fits WGP LDS easily)
    __shared__ float lds[C_NUM * LDS_PITCH];

    const int b    = blockIdx.x >> 6;     // 64 tiles per batch image
    const int tile = blockIdx.x & 63;
    const int s0   = tile * TILE;         // first spatial cell of this tile

    const uint64_t gaddr =
        (uint64_t)(uintptr_t)(x + (size_t)b * (size_t)(C_NUM * WH) + (size_t)s0);

#if __has_builtin(__builtin_amdgcn_tensor_load_to_lds)
    // ---- CDNA5 TDM: async 2D tile DMA global->LDS with hardware LDS padding.
    // Tile: 255 rows x 64 f32, row stride 4096 f32; +1 DWORD pad per 64 DWORDs
    // gives LDS pitch 65 (odd -> bank-conflict free column reads).
    if (threadIdx.x < 32u) {   // one wave issues; TDM ignores EXEC, branch is wave-uniform
        const u32 lds_base = (u32)(uintptr_t)(&lds[0]);   // low 32 bits = LDS offset
        const u32 g0d1 = __builtin_amdgcn_readfirstlane((int)lds_base);
        const u32 g0d2 = __builtin_amdgcn_readfirstlane((int)(u32)gaddr);
        const u32 g0d3 = __builtin_amdgcn_readfirstlane(
            (int)(((u32)(gaddr >> 32) & 0x01FFFFFFu) | (2u << 30)));  // addr[56:32] | type=2

        u32x4 g0 = { 1u,            // count=1 (valid user descriptor), is_restore=0
                     g0d1,          // lds_addr (bytes)
                     g0d2,          // global_addr[31:0]
                     g0d3 };        // global_addr[56:32] | type=2

        i32x8 g1 = {
            (int)((2u << 16)        // data_size = 4 bytes
                | (1u << 20)        // pad_enable
                | (5u << 22)        // pad_interval: 64 DWORDs
                | (0u << 25)),      // pad_amount : 1 DWORD
            (int)(((u32)WH & 0xFFFFu) << 16),                 // tensor_dim0=4096 (lo16)
            (int)((((u32)WH >> 16) & 0xFFFFu) | ((u32)C_NUM << 16)), // dim0 hi16 | tensor_dim1=255
            (int)((u32)TILE << 16),                           // tile_dim0 = 64
            (int)C_NUM,                                       // tile_dim1 = 255, tile_dim2 = 0
            (int)WH,                                          // tensor_dim0_stride = 4096 (lo32)
            0, 0 };

        i32x4 gz4 = {0, 0, 0, 0};   // 2D tensor: groups 2/3 unused
#if __clang_major__ >= 23
        i32x8 gz8 = {0, 0, 0, 0, 0, 0, 0, 0};
        __builtin_amdgcn_tensor_load_to_lds(g0, g1, gz4, gz4, gz8, 0);
#else
        __builtin_amdgcn_tensor_load_to_lds(g0, g1, gz4, gz4, 0);
#endif
        __builtin_amdgcn_s_wait_tensorcnt(0);
    }
#else
    // Fallback: cooperative coalesced load (256B per channel row), manual pitch-65.
    {
        const float* src = x + (size_t)b * (size_t)(C_NUM * WH) + (size_t)s0;
        for (int i = threadIdx.x; i < C_NUM * TILE; i += 256) {
            const int c  = i >> 6;      // channel
            const int sl = i & 63;      // spatial within tile
            lds[c * LDS_PITCH + sl] = src[(size_t)c * WH + sl];
        }
    }
#endif
    __syncthreads();

    // anchors in pixels; the /stride and *stride in the reference cancel.
    const float ax0 = anchors[0], ay0 = anchors[1];
    const float ax1 = anchors[2], ay1 = anchors[3];
    const float ax2 = anchors[4], ay2 = anchors[5];

    float* __restrict__ dst =
        out + (size_t)b * (size_t)(WH * C_NUM) + (size_t)s0 * (size_t)C_NUM;

    // Output tile is 16320 contiguous floats: vectorize as 4080 float4 stores.
    const int nvec = OUT_TILE / 4;            // 4080
    for (int v4 = threadIdx.x; v4 < nvec; v4 += 256) {
        const int j0 = v4 * 4;
        float r[4];
#pragma unroll
        for (int k = 0; k < 4; ++k) {
            const int jj = j0 + k;
            const int d  = jj % D_NUM;        // channel within box descriptor
            const int ra = jj / D_NUM;        // = s_local*A + a
            const int a  = ra % A_NUM;
            const int sl = ra / A_NUM;
            const float val = lds[(a * D_NUM + d) * LDS_PITCH + sl];
            const int s = s0 + sl;            // global cell index, 0..4095
            float res;
            if (d >= 4) {
                res = sigm(val);                                      // conf + classes
            } else if (d == 0) {
                res = (sigm(val) + (float)(s & 63)) * STRIDE_F;       // x + grid_x
            } else if (d == 1) {
                res = (sigm(val) + (float)(s >> 6)) * STRIDE_F;       // y + grid_y
            } else {
                const float anc = (d == 2)
                    ? ((a == 0) ? ax0 : (a == 1) ? ax1 : ax2)
                    : ((a == 0) ? ay0 : (a == 1) ? ay1 : ay2);
                res = __expf(val) * anc;                              // w / h
            }
            r[k] = res;
        }
        float4 stv = make_float4(r[0], r[1], r[2], r[3]);
        *(float4*)(dst + j0) = stv;           // 16B-aligned -> global_store_b128
    }
}

extern "C" void kernel_launch(void* const* d_in, const int* in_sizes, int n_in,
                              void* d_out, int out_size, void* d_ws, size_t ws_size,
                              hipStream_t stream) {
    (void)n_in; (void)out_size; (void)d_ws; (void)ws_size;
    const float* x       = (const float*)d_in[0];
    const float* anchors = (const float*)d_in[1];
    float* out           = (float*)d_out;

    const int B = in_sizes[0] / (C_NUM * WH);     // 32 for the reference shapes
    const int grid = B * (WH / TILE);             // 2048 workgroups
    yolo_v3_head_kernel<<<grid, 256, 0, stream>>>(x, anchors, out);
}